// NeuronMemory_15229954031679
// MI455X (gfx1250) — compile-verified
//
#include <hip/hip_runtime.h>

#define D_MODEL     1024
#define RANK        64
#define N_COMPRESS  16
#define N_KNOWLEDGE 16384
#define TOPK        8
#define TOKENS      (2 * 4096)

typedef float v2f __attribute__((ext_vector_type(2)));
typedef float v8f __attribute__((ext_vector_type(8)));

// ---- CDNA5 async global->LDS copy (ASYNCcnt-tracked, bypasses VGPRs) ------
__device__ __forceinline__ void async_copy_b128(const float* gptr, float* lptr) {
    const unsigned lds_off = (unsigned)(size_t)lptr;          // low 32b of generic LDS ptr
    const unsigned long long ga = (unsigned long long)(size_t)gptr;
    asm volatile("global_load_async_to_lds_b128 %0, %1, off"
                 :: "v"(lds_off), "v"(ga) : "memory");
}
#define WAIT_ASYNC_0() asm volatile("s_wait_asynccnt 0" ::: "memory")

// ---------------------------------------------------------------------------
// Kernel 1: Q[t,r] = sum_n mw[t,n] * (x[t,:] @ cn[n,:,:])[r]
// Block = 16 tokens, 16 waves; wave w owns compress neuron n=w.
// x tile is double-buffered in LDS via global_load_async_to_lds_b128 so the
// next 16x256 chunk streams in while WMMAs run on the current one.
// ---------------------------------------------------------------------------
#define DBLK 256
__global__ __launch_bounds__(512) void nm_q_kernel(
    const float* __restrict__ x, const float* __restrict__ mw,
    const float* __restrict__ cn, float* __restrict__ Q)
{
    __shared__ float x_lds[2][16][DBLK + 4];  // row stride 1040B: 16B-aligned, banks spread
    __shared__ float mw_lds[16][N_COMPRESS];
    __shared__ float q_lds[16][RANK];

    const int t0   = blockIdx.x * 16;
    const int tid  = threadIdx.x;
    const int wave = tid >> 5;
    const int lane = tid & 31;
    const int half = lane >> 4;   // 0: lanes 0-15, 1: lanes 16-31
    const int m    = lane & 15;   // row (A/C) or column (B) within 16
    const int n    = wave;        // compress neuron handled by this wave

    if (tid < 256)
        mw_lds[tid >> 4][tid & 15] = mw[(size_t)(t0 + (tid >> 4)) * N_COMPRESS + (tid & 15)];
    {   // zero the shared Q accumulator (16*64 = 1024 floats)
        float* qf = &q_lds[0][0];
        qf[tid]       = 0.0f;
        qf[tid + 512] = 0.0f;
    }

    const float* cnn = cn + (size_t)n * D_MODEL * RANK;

    v8f acc[4];
#pragma unroll
    for (int rt = 0; rt < 4; ++rt)
#pragma unroll
        for (int i = 0; i < 8; ++i) acc[rt][i] = 0.0f;

    // Issue async stage of the first x chunk: 16 rows x 256 cols = 1024 float4,
    // two b128 copies per thread.
#pragma unroll
    for (int s = 0; s < 2; ++s) {
        const int i   = tid + s * 512;
        const int row = i >> 6;            // 64 float4 per row
        const int c   = (i & 63) * 4;
        async_copy_b128(&x[(size_t)(t0 + row) * D_MODEL + 0 + c], &x_lds[0][row][c]);
    }

    int buf = 0;
    for (int dblk = 0; dblk < D_MODEL; dblk += DBLK) {
        WAIT_ASYNC_0();                     // my async writes for `buf` done
        __syncthreads();                    // everyone's done; prev compute done too
        if (dblk + DBLK < D_MODEL) {        // prefetch next chunk into other buffer
#pragma unroll
            for (int s = 0; s < 2; ++s) {
                const int i   = tid + s * 512;
                const int row = i >> 6;
                const int c   = (i & 63) * 4;
                async_copy_b128(&x[(size_t)(t0 + row) * D_MODEL + dblk + DBLK + c],
                                &x_lds[buf ^ 1][row][c]);
            }
        }

        for (int d0 = 0; d0 < DBLK; d0 += 4) {
            // A fragment: 16x4 f32. VGPR0 = K{0|2}, VGPR1 = K{1|3} by lane half.
            v2f a;
            a[0] = x_lds[buf][m][d0 + 2 * half];
            a[1] = x_lds[buf][m][d0 + 2 * half + 1];
            const float* bp = cnn + (size_t)(dblk + d0) * RANK;
#pragma unroll
            for (int rt = 0; rt < 4; ++rt) {
                // B fragment: 4x16 f32. VGPR0 = K{0|1}, VGPR1 = K{2|3} by half.
                v2f b;
                b[0] = bp[(size_t)half * RANK       + rt * 16 + m];
                b[1] = bp[(size_t)(2 + half) * RANK + rt * 16 + m];
                acc[rt] = __builtin_amdgcn_wmma_f32_16x16x4_f32(
                    false, a, false, b, (short)0, acc[rt], false, false);
            }
        }
        buf ^= 1;
    }

    // Row-scale by routing weight and reduce across the 16 waves (one per n).
#pragma unroll
    for (int rt = 0; rt < 4; ++rt) {
#pragma unroll
        for (int e = 0; e < 8; ++e) {
            const int row = e + 8 * half;          // C/D layout: M = vgpr + 8*half
            atomicAdd(&q_lds[row][rt * 16 + m], mw_lds[row][n] * acc[rt][e]);
        }
    }
    __syncthreads();

    {   // write Q tile coalesced
        const float* qf = &q_lds[0][0];
        Q[(size_t)t0 * RANK + tid]       = qf[tid];
        Q[(size_t)t0 * RANK + 512 + tid] = qf[tid + 512];
    }
}

// ---------------------------------------------------------------------------
// Kernel 2: fused  scores = Q@K^T/8  ->  top-8  ->  softmax  ->  out = w@V
// Block = 16 tokens, 16 waves. Streams N_KNOWLEDGE in 512-wide tiles; the
// 512 MB score matrix never touches HBM. Per-lane running top-8, merged
// wave-wide with a shuffle bitonic merge (wave32).
// ---------------------------------------------------------------------------
__global__ __launch_bounds__(512) void nm_score_topk_kernel(
    const float* __restrict__ Q, const float* __restrict__ Kn,
    const float* __restrict__ Vn, float* __restrict__ out,
    float* __restrict__ out_idx, float* __restrict__ out_w)
{
    const int NT = 512;
    __shared__ float q_lds[16][RANK + 4];   // row stride 272B: 16B-aligned for async b128
    __shared__ float sc_lds[16][NT + 4];
    __shared__ float wv_lds[16][TOPK];
    __shared__ int   wi_lds[16][TOPK];

    const int t0   = blockIdx.x * 16;
    const int tid  = threadIdx.x;
    const int wave = tid >> 5;
    const int lane = tid & 31;
    const int half = lane >> 4;
    const int m    = lane & 15;

    // Async-stage the 16x64 Q tile (one b128 per thread for tid<256).
    if (tid < 256) {
        const int row = tid >> 4;          // 16 float4 per row
        const int c   = (tid & 15) * 4;
        async_copy_b128(&Q[(size_t)(t0 + row) * RANK + c], &q_lds[row][c]);
    }
    WAIT_ASYNC_0();

    float tv[TOPK]; int ti[TOPK];
#pragma unroll
    for (int k = 0; k < TOPK; ++k) { tv[k] = -3.4e38f; ti[k] = 0; }

    const float scale = 0.125f;   // 1/sqrt(RANK)

    for (int jbase = 0; jbase < N_KNOWLEDGE; jbase += NT) {
        __syncthreads();          // protects sc_lds reuse and the q_lds stage
#pragma unroll
        for (int sub = 0; sub < 2; ++sub) {
            const int j0 = jbase + wave * 32 + sub * 16;
            const float* kp = Kn + (size_t)(j0 + m) * RANK;
            if (j0 + NT + m < N_KNOWLEDGE)
                __builtin_prefetch((const void*)(Kn + (size_t)(j0 + NT + m) * RANK), 0, 0);
            v8f acc;
#pragma unroll
            for (int i = 0; i < 8; ++i) acc[i] = 0.0f;
#pragma unroll
            for (int k0 = 0; k0 < RANK; k0 += 4) {
                v2f a;
                a[0] = q_lds[m][k0 + 2 * half];
                a[1] = q_lds[m][k0 + 2 * half + 1];
                v2f b;                       // B[r][j] = Kn[j][r]
                b[0] = kp[k0 + half];
                b[1] = kp[k0 + 2 + half];
                acc = __builtin_amdgcn_wmma_f32_16x16x4_f32(
                    false, a, false, b, (short)0, acc, false, false);
            }
#pragma unroll
            for (int e = 0; e < 8; ++e)
                sc_lds[e + 8 * half][wave * 32 + sub * 16 + m] = acc[e] * scale;
        }
        __syncthreads();
        // Top-8 update: wave w scans token w's 512 scores, 16 per lane.
#pragma unroll
        for (int u = 0; u < NT / 32; ++u) {
            const int c = u * 32 + lane;
            const float v = sc_lds[wave][c];
            if (v > tv[TOPK - 1]) {
                int p = TOPK - 1;
                while (p > 0 && v > tv[p - 1]) {
                    tv[p] = tv[p - 1]; ti[p] = ti[p - 1]; --p;
                }
                tv[p] = v; ti[p] = jbase + c;
            }
        }
    }

    // Wave-wide top-8: xor-butterfly bitonic merge of sorted-desc 8-lists.
#pragma unroll
    for (int off = 16; off >= 1; off >>= 1) {
        float ov[TOPK]; int oi[TOPK];
#pragma unroll
        for (int k = 0; k < TOPK; ++k) {
            ov[k] = __shfl_xor(tv[k], off, 32);
            oi[k] = __shfl_xor(ti[k], off, 32);
        }
        float nv[TOPK]; int ni[TOPK];
#pragma unroll
        for (int k = 0; k < TOPK; ++k) {     // half-cleaner: keep top-8 multiset
            if (tv[k] >= ov[TOPK - 1 - k]) { nv[k] = tv[k];            ni[k] = ti[k]; }
            else                           { nv[k] = ov[TOPK - 1 - k]; ni[k] = oi[TOPK - 1 - k]; }
        }
#pragma unroll
        for (int s = 4; s >= 1; s >>= 1) {   // sort the bitonic 8-list desc
#pragma unroll
            for (int i2 = 0; i2 < TOPK; ++i2) {
                if ((i2 & s) == 0) {
                    const int j2 = i2 | s;
                    if (nv[i2] < nv[j2]) {
                        const float fv = nv[i2]; nv[i2] = nv[j2]; nv[j2] = fv;
                        const int   fi = ni[i2]; ni[i2] = ni[j2]; ni[j2] = fi;
                    }
                }
            }
        }
#pragma unroll
        for (int k = 0; k < TOPK; ++k) { tv[k] = nv[k]; ti[k] = ni[k]; }
    }

    // Softmax over the 8 (all lanes hold identical sorted lists now).
    float ex[TOPK]; float sum = 0.0f;
    const float mx = tv[0];
#pragma unroll
    for (int k = 0; k < TOPK; ++k) { ex[k] = __expf(tv[k] - mx); sum += ex[k]; }
    const float inv = 1.0f / sum;

    if (lane < TOPK) {
        const size_t tok = (size_t)(t0 + wave);
        out_idx[tok * TOPK + lane] = (float)ti[lane];
        out_w  [tok * TOPK + lane] = ex[lane] * inv;
        wv_lds[wave][lane] = ex[lane] * inv;
        wi_lds[wave][lane] = ti[lane];
    }
    __syncthreads();

    // Weighted gather of V rows: out[t,:] = sum_k w_k * V[idx_k,:]
    // Flattened 16 tokens x 256 float4 columns; b128 loads, coalesced stores.
    for (int i = tid; i < 16 * (D_MODEL / 4); i += 512) {
        const int tt = i >> 8;
        const int d  = (i & 255) * 4;
        float4 s = make_float4(0.f, 0.f, 0.f, 0.f);
#pragma unroll
        for (int k = 0; k < TOPK; ++k) {
            const float  w  = wv_lds[tt][k];
            const float4 vv = *(const float4*)&Vn[(size_t)wi_lds[tt][k] * D_MODEL + d];
            s.x += w * vv.x; s.y += w * vv.y; s.z += w * vv.z; s.w += w * vv.w;
        }
        *(float4*)&out[(size_t)(t0 + tt) * D_MODEL + d] = s;
    }
}

// ---------------------------------------------------------------------------
extern "C" void kernel_launch(void* const* d_in, const int* in_sizes, int n_in,
                              void* d_out, int out_size, void* d_ws, size_t ws_size,
                              hipStream_t stream) {
    const float* x  = (const float*)d_in[0];   // [B,S,D]
    const float* mw = (const float*)d_in[1];   // [B,S,16]
    const float* cn = (const float*)d_in[2];   // [16,D,64]
    const float* kK = (const float*)d_in[3];   // [16384,64]
    const float* kV = (const float*)d_in[4];   // [16384,D]

    float* out     = (float*)d_out;                       // [B,S,D]
    float* out_idx = out + (size_t)TOKENS * D_MODEL;      // [B,S,8] (as float)
    float* out_w   = out_idx + (size_t)TOKENS * TOPK;     // [B,S,8]

    float* Q = (float*)d_ws;                              // [TOKENS, RANK] = 2 MB

    dim3 blk(512);
    nm_q_kernel<<<TOKENS / 16, blk, 0, stream>>>(x, mw, cn, Q);
    nm_score_topk_kernel<<<TOKENS / 16, blk, 0, stream>>>(Q, kK, kV, out, out_idx, out_w);
}